// Missing_PromptLearner_61778809585838
// MI455X (gfx1250) — compile-verified
//
#include <hip/hip_runtime.h>
#include <hip/hip_bf16.h>
#include <math.h>

typedef __attribute__((ext_vector_type(2))) float v2f;
typedef __attribute__((ext_vector_type(4))) float v4f;
typedef __attribute__((ext_vector_type(8))) float v8f;

#define DEPTH 12
#define L 16
#define DM 896
#define E 1792
#define H 112
#define HD 56
#define NB 512
#define R 48  /* 3 types * L rows */

__device__ __forceinline__ float gelu_erf(float x) {
  return 0.5f * x * (1.0f + erff(x * 0.70710678118654752f));
}

__device__ __forceinline__ v8f wmma4(v2f a, v2f b, v8f c) {
  return __builtin_amdgcn_wmma_f32_16x16x4_f32(false, a, false, b, (short)0, c,
                                               false, false);
}

// ---------------------------------------------------------------------------
// Gather initial prompts for the 3 missing-types into [48][896] buffers.
// img_tab = [img_c, img_m, img_c]; dep_tab = [dep_c, dep_c, dep_m];
// com_tab = [com_c, com_d, com_i]
// ---------------------------------------------------------------------------
__global__ __launch_bounds__(224) void init_gather(
    const float* __restrict__ img_c, const float* __restrict__ img_m,
    const float* __restrict__ dep_c, const float* __restrict__ dep_m,
    const float* __restrict__ com_c, const float* __restrict__ com_i,
    const float* __restrict__ com_d, float* __restrict__ x0i,
    float* __restrict__ x0d, float* __restrict__ comin) {
  int row = blockIdx.x;  // 0..47
  int t = row >> 4;
  int l = row & 15;
  const float* si = (t == 1) ? img_m : img_c;
  const float* sd = (t == 2) ? dep_m : dep_c;
  const float* sc = (t == 0) ? com_c : ((t == 1) ? com_d : com_i);
  const v4f* si4 = (const v4f*)(si + l * DM);
  const v4f* sd4 = (const v4f*)(sd + l * DM);
  const v4f* sc4 = (const v4f*)(sc + l * DM);
  v4f* di = (v4f*)(x0i + row * DM);
  v4f* dd = (v4f*)(x0d + row * DM);
  v4f* dc = (v4f*)(comin + row * DM);
  int c = threadIdx.x;  // 224 float4 per row
  di[c] = si4[c];
  dd[c] = sd4[c];
  dc[c] = sc4[c];
}

// ---------------------------------------------------------------------------
// LayerNorm stats (mean, rstd) of concat(prev_img, prev_dep) per row.
// ---------------------------------------------------------------------------
__global__ __launch_bounds__(256) void ln_stats(const float* __restrict__ pi,
                                                const float* __restrict__ pd,
                                                float* __restrict__ stats) {
  __shared__ float ssum[256], ssq[256];
  int row = blockIdx.x;
  int tid = threadIdx.x;
  float s = 0.f, q = 0.f;
  for (int i = tid; i < E; i += 256) {
    float x = (i < DM) ? pi[row * DM + i] : pd[row * DM + i - DM];
    s += x;
    q += x * x;
  }
  ssum[tid] = s;
  ssq[tid] = q;
  __syncthreads();
  for (int off = 128; off > 0; off >>= 1) {
    if (tid < off) {
      ssum[tid] += ssum[tid + off];
      ssq[tid] += ssq[tid + off];
    }
    __syncthreads();
  }
  if (tid == 0) {
    float mean = ssum[0] * (1.0f / E);
    float var = ssq[0] * (1.0f / E) - mean * mean;
    stats[row * 2] = mean;
    stats[row * 2 + 1] = rsqrtf(var + 1e-5f);
  }
}

// ---------------------------------------------------------------------------
// GEMM1: GELU(LN(x) @ W1 + b1) -> hbuf[2][48][112]. One wave per (branch,ntile).
// LN affine folded into A-fragment load. K=1792 in steps of 4 (WMMA f32 k4).
// ---------------------------------------------------------------------------
__global__ __launch_bounds__(32) void gemm1_lnmlp(
    const float* __restrict__ pimg, const float* __restrict__ pdep,
    const float* __restrict__ ln_i_g, const float* __restrict__ ln_i_b,
    const float* __restrict__ wi1, const float* __restrict__ bi1,
    const float* __restrict__ ln_d_g, const float* __restrict__ ln_d_b,
    const float* __restrict__ wd1, const float* __restrict__ bd1,
    const float* __restrict__ stats, float* __restrict__ hbuf, int d) {
  const int br = blockIdx.x / 7;
  const int nt = blockIdx.x % 7;
  const float* g = (br ? ln_d_g : ln_i_g) + d * E;
  const float* lb = (br ? ln_d_b : ln_i_b) + d * E;
  const float* W = (br ? wd1 : wi1) + (size_t)d * E * H;
  const float* b1 = (br ? bd1 : bi1) + d * H;
  const int lane = threadIdx.x;
  const int nl = lane & 15;
  const int hi = lane >> 4;
  const int ncol = nt * 16 + nl;
  float bias = b1[ncol];
  v8f acc[3];
#pragma unroll
  for (int mt = 0; mt < 3; ++mt)
#pragma unroll
    for (int v = 0; v < 8; ++v) acc[mt][v] = bias;
  float mean[3], rstd[3];
#pragma unroll
  for (int mt = 0; mt < 3; ++mt) {
    int row = mt * 16 + nl;
    mean[mt] = stats[row * 2];
    rstd[mt] = stats[row * 2 + 1];
  }
  for (int kk = 0; kk < E / 4; ++kk) {
    int klo = kk * 4 + hi * 2;
    v2f gv = *(const v2f*)(g + klo);
    v2f bv = *(const v2f*)(lb + klo);
    v2f bf;
    bf[0] = W[(size_t)klo * H + ncol];
    bf[1] = W[(size_t)(klo + 1) * H + ncol];
    const float* xb = (klo < DM) ? pimg : pdep;
    int kx = (klo < DM) ? klo : klo - DM;
#pragma unroll
    for (int mt = 0; mt < 3; ++mt) {
      int row = mt * 16 + nl;
      v2f xv = *(const v2f*)(xb + row * DM + kx);
      v2f a;
      a[0] = (xv[0] - mean[mt]) * rstd[mt] * gv[0] + bv[0];
      a[1] = (xv[1] - mean[mt]) * rstd[mt] * gv[1] + bv[1];
      acc[mt] = wmma4(a, bf, acc[mt]);
    }
  }
  float* hb = hbuf + br * R * H;
#pragma unroll
  for (int mt = 0; mt < 3; ++mt)
#pragma unroll
    for (int v = 0; v < 8; ++v) {
      int row = mt * 16 + hi * 8 + v;
      hb[row * H + ncol] = gelu_erf(acc[mt][v]);
    }
}

// ---------------------------------------------------------------------------
// GEMM2: hbuf @ W2 + b2 -> ws trajectory [48][896] for this depth.
// One wave per (branch, mtile, ntile). K=112 in steps of 4.
// ---------------------------------------------------------------------------
__global__ __launch_bounds__(32) void gemm2_mlp(
    const float* __restrict__ hbuf, const float* __restrict__ wi2,
    const float* __restrict__ bi2, const float* __restrict__ wd2,
    const float* __restrict__ bd2, float* __restrict__ oimg,
    float* __restrict__ odep, int d) {
  int bx = blockIdx.x;
  int br = bx / 168;
  int rem = bx % 168;
  int mt = rem / 56;
  int nt = rem % 56;
  const float* W = (br ? wd2 : wi2) + (size_t)d * H * DM;
  const float* b2 = (br ? bd2 : bi2) + d * DM;
  const float* hb = hbuf + br * R * H;
  float* out = br ? odep : oimg;
  int lane = threadIdx.x, nl = lane & 15, hi = lane >> 4;
  int ncol = nt * 16 + nl;
  float bias = b2[ncol];
  v8f acc;
#pragma unroll
  for (int v = 0; v < 8; ++v) acc[v] = bias;
  int row = mt * 16 + nl;
  for (int kk = 0; kk < H / 4; ++kk) {
    int klo = kk * 4 + hi * 2;
    v2f a = *(const v2f*)(hb + row * H + klo);
    v2f bf;
    bf[0] = W[klo * DM + ncol];
    bf[1] = W[(klo + 1) * DM + ncol];
    acc = wmma4(a, bf, acc);
  }
#pragma unroll
  for (int v = 0; v < 8; ++v)
    out[(mt * 16 + hi * 8 + v) * DM + ncol] = acc[v];
}

// ---------------------------------------------------------------------------
// Common-prompt MLP stage 1: comin[48][896] @ wc1[896][56] + b -> GELU -> hc.
// N=56 handled with guarded loads in a 64-wide tile set (4 n-tiles).
// ---------------------------------------------------------------------------
__global__ __launch_bounds__(32) void com_gemm1(
    const float* __restrict__ comin, const float* __restrict__ wci1,
    const float* __restrict__ bci1, const float* __restrict__ wcd1,
    const float* __restrict__ bcd1, float* __restrict__ hc) {
  int br = blockIdx.x / 4;
  int nt = blockIdx.x % 4;
  const float* W = br ? wcd1 : wci1;   // [896][56]
  const float* b1 = br ? bcd1 : bci1;  // [56]
  int lane = threadIdx.x, nl = lane & 15, hi = lane >> 4;
  int ncol = nt * 16 + nl;
  bool nok = ncol < HD;
  float bias = nok ? b1[ncol] : 0.f;
  v8f acc[3];
#pragma unroll
  for (int mt = 0; mt < 3; ++mt)
#pragma unroll
    for (int v = 0; v < 8; ++v) acc[mt][v] = bias;
  for (int kk = 0; kk < DM / 4; ++kk) {
    int klo = kk * 4 + hi * 2;
    v2f bf;
    bf[0] = nok ? W[klo * HD + ncol] : 0.f;
    bf[1] = nok ? W[(klo + 1) * HD + ncol] : 0.f;
#pragma unroll
    for (int mt = 0; mt < 3; ++mt) {
      int row = mt * 16 + nl;
      v2f a = *(const v2f*)(comin + row * DM + klo);
      acc[mt] = wmma4(a, bf, acc[mt]);
    }
  }
  if (nok) {
    float* hb = hc + br * R * HD;
#pragma unroll
    for (int mt = 0; mt < 3; ++mt)
#pragma unroll
      for (int v = 0; v < 8; ++v) {
        int row = mt * 16 + hi * 8 + v;
        hb[row * HD + ncol] = gelu_erf(acc[mt][v]);
      }
  }
}

// ---------------------------------------------------------------------------
// Common-prompt MLP stage 2: hc[48][56] @ wc2[56][896] + b -> comI/comD.
// ---------------------------------------------------------------------------
__global__ __launch_bounds__(32) void com_gemm2(
    const float* __restrict__ hc, const float* __restrict__ wci2,
    const float* __restrict__ bci2, const float* __restrict__ wcd2,
    const float* __restrict__ bcd2, float* __restrict__ oI,
    float* __restrict__ oD) {
  int bx = blockIdx.x;
  int br = bx / 168;
  int rem = bx % 168;
  int mt = rem / 56;
  int nt = rem % 56;
  const float* W = br ? wcd2 : wci2;  // [56][896]
  const float* b2 = br ? bcd2 : bci2;
  const float* hb = hc + br * R * HD;
  float* out = br ? oD : oI;
  int lane = threadIdx.x, nl = lane & 15, hi = lane >> 4;
  int ncol = nt * 16 + nl;
  float bias = b2[ncol];
  v8f acc;
#pragma unroll
  for (int v = 0; v < 8; ++v) acc[v] = bias;
  int row = mt * 16 + nl;
  for (int kk = 0; kk < HD / 4; ++kk) {
    int klo = kk * 4 + hi * 2;
    v2f a = *(const v2f*)(hb + row * HD + klo);
    v2f bf;
    bf[0] = W[klo * DM + ncol];
    bf[1] = W[(klo + 1) * DM + ncol];
    acc = wmma4(a, bf, acc);
  }
#pragma unroll
  for (int v = 0; v < 8; ++v)
    out[(mt * 16 + hi * 8 + v) * DM + ncol] = acc[v];
}

// ---------------------------------------------------------------------------
// Broadcast kernels (the bandwidth-bound part): copy the 3-type trajectory
// rows out to all 512 samples. One 224-thread block per output row (float4).
// ---------------------------------------------------------------------------
__global__ __launch_bounds__(224) void bcast0(const int* __restrict__ mt,
                                              const float* __restrict__ traj0,
                                              const float* __restrict__ comp,
                                              float* __restrict__ out) {
  // grid = 512*32 rows; out[b][r][896]
  int b = blockIdx.x >> 5;
  int r = blockIdx.x & 31;
  int t = mt[b];
  const float* src = (r < 16) ? (traj0 + (t * 16 + r) * DM)
                              : (comp + (t * 16 + (r - 16)) * DM);
  const v4f* s4 = (const v4f*)src;
  v4f* d4 = (v4f*)(out + (size_t)blockIdx.x * DM);
  d4[threadIdx.x] = s4[threadIdx.x];
}

__global__ __launch_bounds__(224) void bcast_rest(const int* __restrict__ mt,
                                                  const float* __restrict__ traj,
                                                  float* __restrict__ out) {
  // grid = 11*512*16 rows; out[dd][b][l][896], depth = dd+1
  int idx = blockIdx.x;
  int l = idx & 15;
  int b = (idx >> 4) & 511;
  int dd = idx >> 13;
  int t = mt[b];
  const v4f* s4 =
      (const v4f*)(traj + ((size_t)(dd + 1) * R + t * 16 + l) * DM);
  v4f* d4 = (v4f*)(out + (size_t)idx * DM);
  d4[threadIdx.x] = s4[threadIdx.x];
}

// ---------------------------------------------------------------------------
extern "C" void kernel_launch(void* const* d_in, const int* in_sizes, int n_in,
                              void* d_out, int out_size, void* d_ws,
                              size_t ws_size, hipStream_t stream) {
  const int* mt = (const int*)d_in[0];
  const float* img_c = (const float*)d_in[1];
  const float* img_m = (const float*)d_in[2];
  const float* dep_c = (const float*)d_in[3];
  const float* dep_m = (const float*)d_in[4];
  const float* com_c = (const float*)d_in[5];
  const float* com_i = (const float*)d_in[6];
  const float* com_d = (const float*)d_in[7];
  const float* ln_i_g = (const float*)d_in[8];
  const float* ln_i_b = (const float*)d_in[9];
  const float* wi1 = (const float*)d_in[10];
  const float* bi1 = (const float*)d_in[11];
  const float* wi2 = (const float*)d_in[12];
  const float* bi2 = (const float*)d_in[13];
  const float* ln_d_g = (const float*)d_in[14];
  const float* ln_d_b = (const float*)d_in[15];
  const float* wd1 = (const float*)d_in[16];
  const float* bd1 = (const float*)d_in[17];
  const float* wd2 = (const float*)d_in[18];
  const float* bd2 = (const float*)d_in[19];
  const float* wci1 = (const float*)d_in[20];
  const float* bci1 = (const float*)d_in[21];
  const float* wci2 = (const float*)d_in[22];
  const float* bci2 = (const float*)d_in[23];
  const float* wcd1 = (const float*)d_in[24];
  const float* bcd1 = (const float*)d_in[25];
  const float* wcd2 = (const float*)d_in[26];
  const float* bcd2 = (const float*)d_in[27];

  float* ws = (float*)d_ws;
  const size_t TRJ = (size_t)R * DM;  // 43008 floats per depth-slice
  float* ws_img = ws;                 // [12][48][896]
  float* ws_dep = ws + DEPTH * TRJ;   // [12][48][896]
  float* ws_x0i = ws + 2 * DEPTH * TRJ;
  float* ws_x0d = ws_x0i + TRJ;
  float* ws_comin = ws_x0d + TRJ;
  float* ws_comI = ws_comin + TRJ;
  float* ws_comD = ws_comI + TRJ;
  float* ws_stats = ws_comD + TRJ;          // [48][2]
  float* ws_h = ws_stats + 2 * R;           // [2][48][112]
  float* ws_hc = ws_h + 2 * R * H;          // [2][48][56]

  init_gather<<<48, 224, 0, stream>>>(img_c, img_m, dep_c, dep_m, com_c, com_i,
                                      com_d, ws_x0i, ws_x0d, ws_comin);
  com_gemm1<<<8, 32, 0, stream>>>(ws_comin, wci1, bci1, wcd1, bcd1, ws_hc);
  com_gemm2<<<336, 32, 0, stream>>>(ws_hc, wci2, bci2, wcd2, bcd2, ws_comI,
                                    ws_comD);
  for (int d = 0; d < DEPTH; ++d) {
    const float* pi = (d == 0) ? ws_x0i : ws_img + (size_t)(d - 1) * TRJ;
    const float* pd = (d == 0) ? ws_x0d : ws_dep + (size_t)(d - 1) * TRJ;
    ln_stats<<<48, 256, 0, stream>>>(pi, pd, ws_stats);
    gemm1_lnmlp<<<14, 32, 0, stream>>>(pi, pd, ln_i_g, ln_i_b, wi1, bi1,
                                       ln_d_g, ln_d_b, wd1, bd1, ws_stats,
                                       ws_h, d);
    gemm2_mlp<<<336, 32, 0, stream>>>(ws_h, wi2, bi2, wd2, bd2,
                                      ws_img + (size_t)d * TRJ,
                                      ws_dep + (size_t)d * TRJ, d);
  }

  float* out = (float*)d_out;
  float* o_img0 = out;                      // [512][32][896]
  float* o_imgr = out + 14680064;           // [11][512][16][896]
  float* o_dep0 = out + 95420416;           // [512][32][896]
  float* o_depr = out + 110100480;          // [11][512][16][896]

  bcast0<<<NB * 32, 224, 0, stream>>>(mt, ws_img, ws_comI, o_img0);
  bcast0<<<NB * 32, 224, 0, stream>>>(mt, ws_dep, ws_comD, o_dep0);
  bcast_rest<<<11 * NB * 16, 224, 0, stream>>>(mt, ws_img, o_imgr);
  bcast_rest<<<11 * NB * 16, 224, 0, stream>>>(mt, ws_dep, o_depr);
}